// EdgeNet_21397527069364
// MI455X (gfx1250) — compile-verified
//
#include <hip/hip_runtime.h>
#include <stdint.h>

typedef __attribute__((ext_vector_type(2))) float v2f;
typedef __attribute__((ext_vector_type(8))) float v8f;

// ---------------------------------------------------------------------------
// Kernel 1: B-tile GEMM  (R^T @ X) with V_WMMA_F32_16X16X4_F32
//   R: (N=2048, E=32768) row-major, X: (N, 4).  Output cols colOff..colOff+3
//   of Bws (E x 8).  One wave = 16 edges, K-loop over N in steps of 4.
//   - X staged transposed in LDS -> one ds_load_2addr_b64 pair per 2 WMMAs
//   - B cols 4..15 zeroed with branch-free 0/1 mask (v_pk_mul_f32)
//   - R streamed with nontemporal loads (read-once, don't thrash 192MB L2)
//   - 32-bit element offsets from a uniform base -> saddr-form global loads
// ---------------------------------------------------------------------------
__global__ __launch_bounds__(256) void gemm_wmma(const float* __restrict__ R,
                                                 const float* __restrict__ X,
                                                 float* __restrict__ Bws,
                                                 int N, int E, int colOff) {
  __builtin_assume(N == 2048);   // constant trip count: no remainder loop
  __shared__ float Xt[4 * 2048];  // column-major X: Xt[c*N + k], 32 KB
  for (int k = threadIdx.x; k < N; k += blockDim.x) {
    float4 row = ((const float4*)X)[k];
    Xt[0 * N + k] = row.x;
    Xt[1 * N + k] = row.y;
    Xt[2 * N + k] = row.z;
    Xt[3 * N + k] = row.w;
  }
  __syncthreads();

  const int wid  = threadIdx.x >> 5;
  const int lane = threadIdx.x & 31;
  const int n    = lane & 15;    // = M row for A, = N col for B/D
  const int kh   = lane >> 4;    // K-half select (0: K=0,1  1: K=2,3)
  const int e0   = (blockIdx.x * 8 + wid) * 16;
  const float msk = (n < 4) ? 1.f : 0.f;   // zero pad B cols 4..15, no exec games
  const int   nc  = n & 3;                 // clamped column -> always-valid address

  // 32-bit element offsets from uniform base R (max 2^26 elements)
  uint32_t off0 = (uint32_t)(2 * kh) * (uint32_t)E + (uint32_t)(e0 + n);
  uint32_t off1 = off0 + (uint32_t)E;
  const uint32_t rstep = 4u * (uint32_t)E;   // 4 rows per WMMA step
  const float* xb = &Xt[nc * N + 2 * kh];    // adjacent pair X[r0,nc], X[r0+1,nc]

  v8f acc = {};
#pragma unroll 4
  for (int k4 = 0; k4 < N; k4 += 4) {
    v2f a, b;
    a.x = __builtin_nontemporal_load(R + off0);  // A[m, 2kh]   = R[r0,   e0+m]
    a.y = __builtin_nontemporal_load(R + off1);  // A[m, 2kh+1] = R[r0+1, e0+m]
    float2 xp = *(const float2*)xb;              // ds_load_b64 (8B aligned)
    b.x = msk * xp.x;                            // B[2kh,   n]
    b.y = msk * xp.y;                            // B[2kh+1, n]
    acc = __builtin_amdgcn_wmma_f32_16x16x4_f32(
        false, a, false, b, (short)0, acc, false, false);
    off0 += rstep; off1 += rstep; xb += 4;
  }

  if (n < 4) {
    // D layout: VGPR j holds row (j + 8*kh), col n
    float* o = Bws + (size_t)(e0 + 8 * kh) * 8 + (n + colOff);
#pragma unroll
    for (int j = 0; j < 8; ++j) o[(size_t)j * 8] = acc[j];
  }
}

// ---------------------------------------------------------------------------
// Kernel 2: 8-qubit statevector circuit, one edge per wave32.
//   Flat amp index i = lane*8 + r.  Qubit w <-> flat bit (7-w):
//     q0..q4 -> lane bits 4..0 (masks 16,8,4,2,1), q5..q7 -> r bits 2,1,0.
// ---------------------------------------------------------------------------
struct cf { float x, y; };
struct U3 { cf u00, u01, u10, u11; };

__device__ __forceinline__ U3 make_u3(const float* th, int i) {
  float t = th[i], p = th[i + 1], l = th[i + 2];
  float st, ct, sp, cp, sl, cl;
  sincosf(0.5f * t, &st, &ct);
  sincosf(p, &sp, &cp);
  sincosf(l, &sl, &cl);
  U3 u;
  u.u00 = { ct, 0.f };
  u.u01 = { -cl * st, -sl * st };
  u.u10 = { cp * st, sp * st };
  u.u11 = { (cp * cl - sp * sl) * ct, (cp * sl + sp * cl) * ct };
  return u;
}

// d = u0*(ar,ai) + u1*(br,bi)
__device__ __forceinline__ void capply(float& dr, float& di,
                                       float u0x, float u0y, float ar, float ai,
                                       float u1x, float u1y, float br, float bi) {
  dr = u0x * ar - u0y * ai + u1x * br - u1y * bi;
  di = u0x * ai + u0y * ar + u1x * bi + u1y * br;
}

template <int P>
__device__ __forceinline__ void u3_local(float (&re)[8], float (&im)[8], U3 u) {
#pragma unroll
  for (int r0 = 0; r0 < 8; ++r0) {
    if (r0 & (1 << P)) continue;
    const int r1 = r0 | (1 << P);
    float ar = re[r0], ai = im[r0], br = re[r1], bi = im[r1];
    capply(re[r0], im[r0], u.u00.x, u.u00.y, ar, ai, u.u01.x, u.u01.y, br, bi);
    capply(re[r1], im[r1], u.u10.x, u.u10.y, ar, ai, u.u11.x, u.u11.y, br, bi);
  }
}

__device__ __forceinline__ void u3_xlane(float (&re)[8], float (&im)[8], U3 u,
                                         int lane, int mask) {
  const bool hi = (lane & mask) != 0;
  float u0x = hi ? u.u10.x : u.u00.x, u0y = hi ? u.u10.y : u.u00.y;
  float u1x = hi ? u.u11.x : u.u01.x, u1y = hi ? u.u11.y : u.u01.y;
#pragma unroll
  for (int r = 0; r < 8; ++r) {
    float ore = __shfl_xor(re[r], mask, 32);
    float oim = __shfl_xor(im[r], mask, 32);
    float ar = hi ? ore : re[r], ai = hi ? oim : im[r];   // the |0> amp
    float br = hi ? re[r] : ore, bi = hi ? im[r] : oim;   // the |1> amp
    capply(re[r], im[r], u0x, u0y, ar, ai, u1x, u1y, br, bi);
  }
}

__device__ __forceinline__ void cx_xlane(float (&re)[8], float (&im)[8],
                                         int lane, int cmask, int tmask) {
  const bool c = (lane & cmask) != 0;
#pragma unroll
  for (int r = 0; r < 8; ++r) {
    float ore = __shfl_xor(re[r], tmask, 32);
    float oim = __shfl_xor(im[r], tmask, 32);
    re[r] = c ? ore : re[r];
    im[r] = c ? oim : im[r];
  }
}

template <int P>
__device__ __forceinline__ void cx_lane_local(float (&re)[8], float (&im)[8],
                                              int lane, int cmask) {
  const bool c = (lane & cmask) != 0;
  float nr[8], ni[8];
#pragma unroll
  for (int r = 0; r < 8; ++r) {
    nr[r] = c ? re[r ^ (1 << P)] : re[r];
    ni[r] = c ? im[r ^ (1 << P)] : im[r];
  }
#pragma unroll
  for (int r = 0; r < 8; ++r) { re[r] = nr[r]; im[r] = ni[r]; }
}

__device__ __forceinline__ void fswap(float& a, float& b) { float t = a; a = b; b = t; }

__global__ __launch_bounds__(256) void circuit_kernel(const float* __restrict__ Bws,
                                                      const float* __restrict__ th,
                                                      float* __restrict__ out, int E) {
  const int lane = threadIdx.x & 31;
  const int wid  = threadIdx.x >> 5;
  const int e    = blockIdx.x * (blockDim.x >> 5) + wid;
  if (e >= E) return;

  float cs[8], sn[8];
#pragma unroll
  for (int w = 0; w < 8; ++w) {
    float h = 0.5f * Bws[(size_t)e * 8 + w];
    sincosf(h, &sn[w], &cs[w]);
  }
  float lf = 1.f;   // product of the 5 lane-bit qubit factors
#pragma unroll
  for (int w = 0; w < 5; ++w) {
    int bit = (lane >> (4 - w)) & 1;
    lf *= bit ? sn[w] : cs[w];
  }
  float re[8], im[8];
#pragma unroll
  for (int r = 0; r < 8; ++r) {
    float f = lf;
    f *= (r & 4) ? sn[5] : cs[5];
    f *= (r & 2) ? sn[6] : cs[6];
    f *= (r & 1) ? sn[7] : cs[7];
    re[r] = f; im[r] = 0.f;
  }

  u3_xlane(re, im, make_u3(th, 0), lane, 16);   // u3 q0
  u3_xlane(re, im, make_u3(th, 3), lane, 8);    // u3 q1
  cx_xlane(re, im, lane, 16, 8);                // cx(0,1)
  u3_xlane(re, im, make_u3(th, 6), lane, 4);    // u3 q2
  u3_xlane(re, im, make_u3(th, 9), lane, 2);    // u3 q3
  cx_xlane(re, im, lane, 2, 4);                 // cx(3,2)
  u3_xlane(re, im, make_u3(th, 12), lane, 1);   // u3 q4
  u3_local<2>(re, im, make_u3(th, 15));         // u3 q5
  cx_lane_local<2>(re, im, lane, 1);            // cx(4,5)
  u3_local<1>(re, im, make_u3(th, 18));         // u3 q6
  u3_local<0>(re, im, make_u3(th, 21));         // u3 q7
  fswap(re[1], re[3]); fswap(im[1], im[3]);     // cx(7,6): r-bit0 controls r-bit1
  fswap(re[5], re[7]); fswap(im[5], im[7]);
  u3_xlane(re, im, make_u3(th, 24), lane, 8);   // u3 q1
  u3_xlane(re, im, make_u3(th, 27), lane, 4);   // u3 q2
  cx_xlane(re, im, lane, 8, 4);                 // cx(1,2)
  u3_local<2>(re, im, make_u3(th, 30));         // u3 q5
  u3_local<1>(re, im, make_u3(th, 33));         // u3 q6
  fswap(re[2], re[6]); fswap(im[2], im[6]);     // cx(6,5): r-bit1 controls r-bit2
  fswap(re[3], re[7]); fswap(im[3], im[7]);
  u3_xlane(re, im, make_u3(th, 36), lane, 4);   // u3 q2
  u3_local<2>(re, im, make_u3(th, 39));         // u3 q5
  cx_lane_local<2>(re, im, lane, 4);            // cx(2,5)
  u3_local<2>(re, im, make_u3(th, 42));         // u3 q5

  // out[e] = P(q5 = 1) = sum |amp|^2 over r-bit2 = 1, reduced over lanes
  float p = 0.f;
#pragma unroll
  for (int r = 4; r < 8; ++r) p += re[r] * re[r] + im[r] * im[r];
#pragma unroll
  for (int off = 16; off > 0; off >>= 1) p += __shfl_xor(p, off, 32);
  if (lane == 0) out[e] = p;
}

extern "C" void kernel_launch(void* const* d_in, const int* in_sizes, int n_in,
                              void* d_out, int out_size, void* d_ws, size_t ws_size,
                              hipStream_t stream) {
  const float* X  = (const float*)d_in[0];   // (N, 4)
  const float* Ri = (const float*)d_in[1];   // (N, E)
  const float* Ro = (const float*)d_in[2];   // (N, E)
  const float* th = (const float*)d_in[3];   // (45,)
  float* out = (float*)d_out;                // (E,)
  const int N = in_sizes[0] / 4;             // 2048
  const int E = out_size;                    // 32768
  float* Bws = (float*)d_ws;                 // (E, 8) scratch, fully written below

  dim3 blk(256);
  dim3 g1(E / 128);                          // 8 waves x 16 edges per block
  gemm_wmma<<<g1, blk, 0, stream>>>(Ro, X, Bws, N, E, 0);  // bo -> cols 0..3
  gemm_wmma<<<g1, blk, 0, stream>>>(Ri, X, Bws, N, E, 4);  // bi -> cols 4..7
  dim3 g2((E + 7) / 8);                      // one edge per wave, 8 waves/block
  circuit_kernel<<<g2, blk, 0, stream>>>(Bws, th, out, E);
}